// SNNDecode_63479616635538
// MI455X (gfx1250) — compile-verified
//
#include <hip/hip_runtime.h>

typedef float v2f __attribute__((ext_vector_type(2)));
typedef float v4f __attribute__((ext_vector_type(4)));
typedef float v8f __attribute__((ext_vector_type(8)));

#define SEQ     2048
#define BATCH   64
#define HIDDEN  512
#define OUT     128
#define NSEG    8
#define SEGLEN  (SEQ / NSEG)      /* 256 */
#define CHUNK   16
#define NCHUNK  (SEGLEN / CHUNK)  /* 16 */
#define WPITCH  516               /* 512 + 4 pad dwords: bank stride 4, 16B aligned rows */
#define ZPITCH  132               /* 128 + 4 pad dwords */
#define VOLT_ELEMS ((size_t)SEQ * BATCH * OUT)

#define XSH_OFF (OUT * WPITCH)               /* 66048 floats */
#define ZT_OFF  (XSH_OFF + CHUNK * WPITCH)   /* 74304 floats */
#define LDS_FLOATS (ZT_OFF + CHUNK * ZPITCH) /* 76416 floats = 305664 B <= 320K */

__device__ __forceinline__ void load_taus(const float* ts_p, const float* tm_p,
                                          float& dts, float& dtm) {
    float ts = fminf(fmaxf(ts_p[0], 0.0f), 1.0f);
    float tm = fminf(fmaxf(tm_p[0], 0.0f), 1.0f);
    dts = 0.001f * ts;
    dtm = 0.001f * tm;
}

// ---------------------------------------------------------------------------
// Kernel 1: per (batch, segment): z = x @ W^T via V_WMMA_F32_16X16X4_F32,
// zero-initial-state LI scan over the segment, voltages -> d_out,
// segment-final (v,i) -> cseg workspace.
// Software-pipelined x staging; 2 block barriers per chunk; dual WMMA
// accumulators break the accumulator dependency chain in the K loop.
// ---------------------------------------------------------------------------
__global__ __launch_bounds__(256, 1)
void snn_seg_kernel(const float* __restrict__ x, const float* __restrict__ W,
                    const float* __restrict__ tau_syn_p, const float* __restrict__ tau_mem_p,
                    float* __restrict__ out, float* __restrict__ cseg) {
    __shared__ float smem[LDS_FLOATS];
    float* wsh = smem;            // [128][516]
    float* xsh = smem + XSH_OFF;  // [16][516]
    float* zt  = smem + ZT_OFF;   // [16][132]  per-wave 16-column z slices

    const int g    = blockIdx.x;
    const int b    = blockIdx.y;
    const int tid  = threadIdx.x;
    const int lane = tid & 31;
    const int wave = tid >> 5;
    const int n0   = wave * 16;     // this wave's 16 output columns
    const int mn   = lane & 15;     // A row (timestep) / B row (out col) index
    const int h    = lane >> 4;     // half-wave selects K pair (ISA 7.12.2 f32 layout)

    float dts, dtm;
    load_taus(tau_syn_p, tau_mem_p, dts, dtm);

    const size_t xbase = ((size_t)b * SEQ + (size_t)g * SEGLEN) * HIDDEN;

    // Prefetch chunk 0 into registers (in flight during W preload)
    v4f xr[8];
    {
        const v4f* xv = (const v4f*)(x + xbase);
        #pragma unroll
        for (int j = 0; j < 8; ++j) xr[j] = xv[tid + 256 * j];
    }

    // Preload W [128][512] -> LDS padded rows (coalesced b128 reads)
    {
        const v4f* Wv = (const v4f*)W;
        for (int i = tid; i < OUT * (HIDDEN / 4); i += 256) {
            int row = i >> 7;
            int c4  = i & 127;
            *(v4f*)&wsh[row * WPITCH + c4 * 4] = Wv[i];
        }
    }

    const float* aRow = &xsh[mn * WPITCH + 2 * h];
    const float* bRow = &wsh[(n0 + mn) * WPITCH + 2 * h];

    float v = 0.0f, cur = 0.0f;   // scan state (lanes 0..15, column n0+mn)

    for (int ch = 0; ch < NCHUNK; ++ch) {
        __syncthreads();  // all waves done reading xsh of previous chunk
        // Commit prefetched chunk to LDS
        #pragma unroll
        for (int j = 0; j < 8; ++j) {
            int idx = tid + 256 * j;
            *(v4f*)&xsh[(idx >> 7) * WPITCH + (idx & 127) * 4] = xr[j];
        }
        __syncthreads();  // xsh ready

        // Launch next chunk's global reads; they overlap the WMMA loop below
        if (ch + 1 < NCHUNK) {
            const v4f* xn = (const v4f*)(x + xbase + (size_t)(ch + 1) * CHUNK * HIDDEN);
            #pragma unroll
            for (int j = 0; j < 8; ++j) xr[j] = xn[tid + 256 * j];
        }

        // 16x16 z tile, K = 512 -> 128 fp32 WMMAs on two independent
        // accumulators (even/odd K blocks) to avoid back-to-back RAW stalls.
        v8f acc0 = {0.f, 0.f, 0.f, 0.f, 0.f, 0.f, 0.f, 0.f};
        v8f acc1 = {0.f, 0.f, 0.f, 0.f, 0.f, 0.f, 0.f, 0.f};
        #pragma unroll 4
        for (int k = 0; k < HIDDEN; k += 8) {
            v2f a0 = *(const v2f*)&aRow[k];
            v2f b0 = *(const v2f*)&bRow[k];
            v2f a1 = *(const v2f*)&aRow[k + 4];
            v2f b1 = *(const v2f*)&bRow[k + 4];
            acc0 = __builtin_amdgcn_wmma_f32_16x16x4_f32(false, a0, false, b0,
                                                         (short)0, acc0, false, false);
            acc1 = __builtin_amdgcn_wmma_f32_16x16x4_f32(false, a1, false, b1,
                                                         (short)0, acc1, false, false);
        }

        // D layout: lane L reg r -> (M = r + 8h, N = L&15). Wave-local spill.
        #pragma unroll
        for (int r = 0; r < 8; ++r) {
            zt[(r + 8 * h) * ZPITCH + n0 + mn] = acc0[r] + acc1[r];
        }
        // Wave-local LDS handoff: in-order DS pipe, drain this wave's stores.
        asm volatile("s_wait_dscnt 0" ::: "memory");

        // Serial LI scan; voltages stored straight to global (16 floats/row/wave)
        if (lane < 16) {
            const int o = n0 + mn;
            float* orow = out + ((size_t)(g * SEGLEN + ch * CHUNK) * BATCH + b) * OUT + o;
            #pragma unroll
            for (int t = 0; t < CHUNK; ++t) {
                float z  = zt[t * ZPITCH + o];
                float vn = v + dtm * (cur - v);
                cur = cur - dts * cur + z;
                v = vn;
                orow[(size_t)t * BATCH * OUT] = vn;
            }
        }
    }

    // Segment forced-response final state -> workspace
    if (lane < 16) {
        const size_t e = ((size_t)g * BATCH + b) * OUT + (n0 + mn);
        cseg[2 * e]     = v;
        cseg[2 * e + 1] = cur;
    }
}

// ---------------------------------------------------------------------------
// Kernel 2: per (b,o) chain, sequentially combine the 8 segment responses:
// s_{g+1} = A^SEGLEN * s_g + c_g. Emits each segment's true initial state and
// the exact final (v_f, i_f) into d_out.
// ---------------------------------------------------------------------------
__global__ __launch_bounds__(256)
void snn_combine_kernel(const float* __restrict__ tau_syn_p, const float* __restrict__ tau_mem_p,
                        const float* __restrict__ cseg, float* __restrict__ init,
                        float* __restrict__ out) {
    const int idx = blockIdx.x * 256 + threadIdx.x;   // (b,o) flat, 8192 total
    if (idx >= BATCH * OUT) return;
    float dts, dtm;
    load_taus(tau_syn_p, tau_mem_p, dts, dtm);

    float v = 0.0f, cur = 0.0f;
    for (int g = 0; g < NSEG; ++g) {
        const size_t e = (size_t)g * BATCH * OUT + idx;
        init[2 * e]     = v;
        init[2 * e + 1] = cur;
        #pragma unroll 4
        for (int t = 0; t < SEGLEN; ++t) {   // homogeneous propagation A^SEGLEN
            float vn = v + dtm * (cur - v);
            cur = cur - dts * cur;
            v = vn;
        }
        v   += cseg[2 * e];
        cur += cseg[2 * e + 1];
    }
    out[VOLT_ELEMS + idx]               = v;    // v_f
    out[VOLT_ELEMS + BATCH * OUT + idx] = cur;  // i_f
}

// ---------------------------------------------------------------------------
// Kernel 3: add homogeneous correction A^(t+1) * s0 to voltages of segments
// 1..NSEG-1 (segment 0 has zero initial state). Coalesced RMW, no z re-read.
// ---------------------------------------------------------------------------
__global__ __launch_bounds__(128)
void snn_correct_kernel(const float* __restrict__ tau_syn_p, const float* __restrict__ tau_mem_p,
                        const float* __restrict__ init, float* __restrict__ out) {
    const int o = threadIdx.x;
    const int b = blockIdx.x;
    const int g = blockIdx.y + 1;
    float dts, dtm;
    load_taus(tau_syn_p, tau_mem_p, dts, dtm);

    const size_t e = ((size_t)g * BATCH + b) * OUT + o;
    float v = init[2 * e], cur = init[2 * e + 1];
    #pragma unroll 4
    for (int t = 0; t < SEGLEN; ++t) {
        float vn = v + dtm * (cur - v);
        cur = cur - dts * cur;
        v = vn;
        const size_t oi = ((size_t)(g * SEGLEN + t) * BATCH + b) * OUT + o;
        out[oi] += v;
    }
}

// ---------------------------------------------------------------------------
extern "C" void kernel_launch(void* const* d_in, const int* in_sizes, int n_in,
                              void* d_out, int out_size, void* d_ws, size_t ws_size,
                              hipStream_t stream) {
    const float* x       = (const float*)d_in[0];
    const float* W       = (const float*)d_in[1];
    const float* tau_syn = (const float*)d_in[2];
    const float* tau_mem = (const float*)d_in[3];
    float* out  = (float*)d_out;
    float* cseg = (float*)d_ws;                              // [NSEG][B][O][2]
    float* init = cseg + (size_t)2 * NSEG * BATCH * OUT;     // [NSEG][B][O][2]

    snn_seg_kernel<<<dim3(NSEG, BATCH), 256, 0, stream>>>(x, W, tau_syn, tau_mem, out, cseg);
    snn_combine_kernel<<<dim3((BATCH * OUT) / 256), 256, 0, stream>>>(tau_syn, tau_mem,
                                                                      cseg, init, out);
    snn_correct_kernel<<<dim3(BATCH, NSEG - 1), 128, 0, stream>>>(tau_syn, tau_mem, init, out);
}